// ClusterModel_60000693125187
// MI455X (gfx1250) — compile-verified
//
#include <hip/hip_runtime.h>
#include <math.h>

// ---------------------------------------------------------------------------
// k-means (Lloyd) for N x 128 f32 points, K=8 clusters, on MI455X (gfx1250).
// Distance GEMM X @ C^T runs on V_WMMA_F32_16X16X4_F32 (full f32 precision,
// required because distances feed an argmin). Memory-bound problem: X = 64MB
// fits in the 192MB L2, so the 100-iteration loop is L2-resident.
// Device-side convergence flag (graph-capture safe, deterministic per call).
// ---------------------------------------------------------------------------

typedef __attribute__((ext_vector_type(2))) float v2f;
typedef __attribute__((ext_vector_type(8))) float v8f;

static constexpr int kD = 128;   // feature dim
static constexpr int kK = 8;     // clusters
static constexpr int kIters = 100;
static constexpr float kTol = 1e-6f;

// workspace layout, in 32-bit words
static constexpr int WS_CENTERS = 0;     // kK*kD f32
static constexpr int WS_SUMS    = 1024;  // kK*kD f32
static constexpr int WS_CNORM   = 2048;  // kK f32
static constexpr int WS_COUNTS  = 2056;  // kK u32
static constexpr int WS_FLAG    = 2064;  // 1 u32
static constexpr int WS_MINB    = 2072;  // kK u32 (float bits)
static constexpr int WS_PRESENT = 2080;  // kK u32

// ---------------------------------------------------------------------------
// Shared helper: one block covers 128 rows (8 waves x 16 rows). Each wave
// computes its 16x8 dot tile with WMMA f32 16x16x4, accumulating over D=128
// in 32 chunks of K=4. Also computes per-row ||x||^2 on the fly.
// A frag (16x4): lane l, vgpr v -> M=l%16, Kidx=v+2*(l/16)   (float2 / lane)
// B frag (4x16): lane l, vgpr v -> N=l%16, Kidx=v+2*(l/16)   (float2 / lane)
// D tile (16x16): lane l, vgpr v -> M=v+8*(l/16), N=l%16
// ---------------------------------------------------------------------------
__device__ __forceinline__ void compute_tile(
    const float* __restrict__ X, const float* __restrict__ centers,
    int rowBase, float (*ct)[130], float (*wd)[16][9], float (*xn)[16]) {
  const int t = threadIdx.x;
  // stage centers into LDS, zero-padded to 16 rows, pitch 130 (bank spread)
  for (int i = t; i < 16 * 130; i += 256) {
    const int r = i / 130, c = i % 130;
    ct[r][c] = (r < kK && c < kD) ? centers[r * kD + c] : 0.0f;
  }
  __syncthreads();

  const int lane = t & 31;
  const int w    = t >> 5;         // wave id 0..7 -> rows [w*16, w*16+16)
  const int half = lane >> 4;      // 0 or 1
  const int m    = lane & 15;      // row (A) / col (B)

  const float* ap = X + (size_t)(rowBase + w * 16 + m) * kD + 2 * half;
  const float* bp = &ct[m][2 * half];

  v8f acc = {0.f, 0.f, 0.f, 0.f, 0.f, 0.f, 0.f, 0.f};
  float xsq = 0.0f;
  for (int kc = 0; kc < kD / 4; ++kc) {
    v2f a = *(const v2f*)(ap + 4 * kc);   // global, L2-resident
    v2f b = *(const v2f*)(bp + 4 * kc);   // LDS
    xsq += a.x * a.x + a.y * a.y;
    // D = A(16x4) * B(4x16) + C   -- v_wmma_f32_16x16x4_f32
    acc = __builtin_amdgcn_wmma_f32_16x16x4_f32(
        false, a, false, b, (short)0, acc, false, false);
  }
  // lane r and lane r+16 each hold half of ||x_r||^2
  xsq += __shfl_xor(xsq, 16, 32);
  if (lane < 16) xn[w][lane] = xsq;
  if (m < kK) {
#pragma unroll
    for (int v = 0; v < 8; ++v) wd[w][v + 8 * half][m] = acc[v];
  }
  __syncthreads();
}

// ---------------------------------------------------------------------------
__global__ __launch_bounds__(256) void k_init(const float* __restrict__ X,
                                              float* ws, int N) {
  float* centers = ws + WS_CENTERS;
  float* sums    = ws + WS_SUMS;
  float* cnorm   = ws + WS_CNORM;
  unsigned* cnts = (unsigned*)ws + WS_COUNTS;
  unsigned* flag = (unsigned*)ws + WS_FLAG;
  unsigned* minb = (unsigned*)ws + WS_MINB;
  unsigned* pres = (unsigned*)ws + WS_PRESENT;
  const int t = threadIdx.x;
  for (int i = t; i < kK * kD; i += 256) {
    const int c = i / kD, d = i % kD;
    const int src = c * (N / kK);  // deterministic spread init rows
    centers[i] = X[(size_t)src * kD + d];
    sums[i] = 0.0f;
  }
  if (t < kK) { cnts[t] = 0u; minb[t] = 0x7F800000u; pres[t] = 0u; }
  if (t == 0) *flag = 0u;
  __syncthreads();
  if (t < kK) {
    float s = 0.0f;
    for (int d = 0; d < kD; ++d) { float v = centers[t * kD + d]; s += v * v; }
    cnorm[t] = s;
  }
}

// ---------------------------------------------------------------------------
__global__ __launch_bounds__(256) void k_assign(const float* __restrict__ X,
                                                float* ws, int N) {
  unsigned* flag = (unsigned*)ws + WS_FLAG;
  if (*flag) return;  // converged: whole launch is a no-op (uniform)

  __shared__ float ct[16][130];
  __shared__ float wd[8][16][9];
  __shared__ float xn[8][16];
  __shared__ int   choice[128];
  __shared__ float ssum[2][kK * kD];

  const float* centers = ws + WS_CENTERS;
  const float* cnorm   = ws + WS_CNORM;
  float*    gsums  = ws + WS_SUMS;
  unsigned* gcnts  = (unsigned*)ws + WS_COUNTS;

  const int t = threadIdx.x;
  const int rowBase = blockIdx.x * 128;

  compute_tile(X, centers, rowBase, ct, wd, xn);

  if (t < 128) {  // one thread per row: argmin over 8 centers
    const int w = t >> 4, r = t & 15;
    const float xs = xn[w][r];
    int bc = 0; float bd = 3.4e38f;
#pragma unroll
    for (int c = 0; c < kK; ++c) {
      const float dv = fmaxf(xs + cnorm[c] - 2.0f * wd[w][r][c], 0.0f);
      if (dv < bd) { bd = dv; bc = c; }
    }
    choice[t] = bc;
  }
  for (int i = t; i < 2 * kK * kD; i += 256) ((float*)ssum)[i] = 0.0f;
  __syncthreads();

  // cluster-sum accumulation: thread (copy,d) exclusively owns ssum[copy][*][d]
  {
    const int d = t & 127, cp = t >> 7;
    for (int r = cp * 64; r < cp * 64 + 64; ++r) {
      const int c = choice[r];
      ssum[cp][c * kD + d] += X[(size_t)(rowBase + r) * kD + d];
    }
  }
  __syncthreads();
  for (int i = t; i < kK * kD; i += 256)
    atomicAdd(&gsums[i], ssum[0][i] + ssum[1][i]);
  if (t < kK) {
    unsigned cnt = 0;
    for (int r = 0; r < 128; ++r) cnt += (choice[r] == t);
    if (cnt) atomicAdd(&gcnts[t], cnt);
  }
}

// ---------------------------------------------------------------------------
__global__ __launch_bounds__(256) void k_update(float* ws) {
  unsigned* flag = (unsigned*)ws + WS_FLAG;
  if (*flag) return;

  __shared__ float red[kK * kD];
  __shared__ float red2[kK * kD];
  __shared__ float spart[kK];

  float*    centers = ws + WS_CENTERS;
  float*    sums    = ws + WS_SUMS;
  float*    cnorm   = ws + WS_CNORM;
  unsigned* cnts    = (unsigned*)ws + WS_COUNTS;
  const int t = threadIdx.x;

  for (int i = t; i < kK * kD; i += 256) {
    const int c = i / kD;
    const unsigned cnt = cnts[c];
    const float oldv = centers[i];
    const float nc = cnt ? sums[i] / (float)cnt : oldv;  // empty: keep old
    centers[i] = nc;
    const float dl = nc - oldv;
    red[i]  = dl * dl;
    red2[i] = nc * nc;
    sums[i] = 0.0f;  // ready for next iteration
  }
  __syncthreads();
  if (t < kK) {
    float s = 0.0f, s2 = 0.0f;
    for (int d = 0; d < kD; ++d) { s += red[t * kD + d]; s2 += red2[t * kD + d]; }
    spart[t] = sqrtf(s);
    cnorm[t] = s2;
    cnts[t]  = 0u;
  }
  __syncthreads();
  if (t == 0) {
    float sh = 0.0f;
    for (int c = 0; c < kK; ++c) sh += spart[c];
    if (sh * sh < kTol) *flag = 1u;
  }
}

// ---------------------------------------------------------------------------
__global__ __launch_bounds__(256) void k_final_assign(
    const float* __restrict__ X, float* ws, float* __restrict__ out, int N) {
  __shared__ float ct[16][130];
  __shared__ float wd[8][16][9];
  __shared__ float xn[8][16];

  const float* centers = ws + WS_CENTERS;
  const float* cnorm   = ws + WS_CNORM;
  unsigned* minb = (unsigned*)ws + WS_MINB;
  unsigned* pres = (unsigned*)ws + WS_PRESENT;

  const int t = threadIdx.x;
  const int rowBase = blockIdx.x * 128;
  const size_t offDist = (size_t)N + kK * kD;

  compute_tile(X, centers, rowBase, ct, wd, xn);

  if (t < 128) {
    const int w = t >> 4, r = t & 15;
    const int row = rowBase + t;
    const float xs = xn[w][r];
    float dist[kK];
    int bc = 0; float bd = 3.4e38f;
#pragma unroll
    for (int c = 0; c < kK; ++c) {
      const float dv = fmaxf(xs + cnorm[c] - 2.0f * wd[w][r][c], 0.0f);
      dist[c] = dv;
      out[offDist + (size_t)row * kK + c] = dv;
      if (dv < bd) { bd = dv; bc = c; }
    }
    out[row] = (float)bc;  // cluster id
    atomicOr(&pres[bc], 1u);
    // dist >= 0 so float bits order like uints
    atomicMin(&minb[bc], __float_as_uint(dist[bc]));
  }
}

// ---------------------------------------------------------------------------
__global__ __launch_bounds__(256) void k_finalize(float* ws,
                                                  float* __restrict__ out,
                                                  int N) {
  const unsigned* minb = (const unsigned*)ws + WS_MINB;
  const unsigned* pres = (const unsigned*)ws + WS_PRESENT;
  const int n = blockIdx.x * 256 + threadIdx.x;
  const size_t offDist = (size_t)N + kK * kD;
  const size_t offIsc  = offDist + (size_t)N * kK;
  if (n < N) {
    int isc = 0;
#pragma unroll
    for (int c = 0; c < kK; ++c) {
      const float dv = out[offDist + (size_t)n * kK + c];
      if (pres[c] && (__float_as_uint(dv) == minb[c])) isc = 1;
    }
    out[offIsc + n] = (float)isc;
  }
  if (blockIdx.x == 0) {  // copy final centers into output slot 2
    const float* centers = ws + WS_CENTERS;
    for (int i = threadIdx.x; i < kK * kD; i += 256) out[N + i] = centers[i];
  }
}

// ---------------------------------------------------------------------------
extern "C" void kernel_launch(void* const* d_in, const int* in_sizes, int n_in,
                              void* d_out, int out_size, void* d_ws,
                              size_t ws_size, hipStream_t stream) {
  const float* X = (const float*)d_in[0];
  const int N = in_sizes[0] / kD;     // 131072
  float* ws  = (float*)d_ws;
  float* out = (float*)d_out;
  const int nb = N / 128;             // 1024 blocks, 8 waves each

  k_init<<<1, 256, 0, stream>>>(X, ws, N);
  for (int it = 0; it < kIters; ++it) {
    k_assign<<<nb, 256, 0, stream>>>(X, ws, N);
    k_update<<<1, 256, 0, stream>>>(ws);
  }
  k_final_assign<<<nb, 256, 0, stream>>>(X, ws, out, N);
  k_finalize<<<(N + 255) / 256, 256, 0, stream>>>(ws, out, N);
}